// ObjectAwareCrossAttention_2559800508520
// MI455X (gfx1250) — compile-verified
//
#include <hip/hip_runtime.h>
#include <hip/hip_bf16.h>
#include <math.h>

// ---------------- problem constants ----------------
#define BB      8
#define CC      512
#define L1C     1024          // H*W
#define ENC     1024
#define L2C     64
#define NHEAD   8
#define DHEAD   64            // CC / NHEAD
#define GROUPS  32
#define EPSV    1e-5f
#define KC      32            // GEMM K-slab staged in LDS

typedef __attribute__((ext_vector_type(2))) float v2f;
typedef __attribute__((ext_vector_type(8))) float v8f;

// ---- gfx1250 async global->LDS path (guarded; falls back clean) ----
#if defined(__has_builtin)
#if __has_builtin(__builtin_amdgcn_global_load_async_to_lds_b128) && \
    __has_builtin(__builtin_amdgcn_s_wait_asynccnt)
#define USE_ASYNC_LDS 1
#endif
#endif

// exact element type from the builtin's signature: int __attribute__((vector_size(16)))
typedef int v4i_vs __attribute__((vector_size(16)));
typedef __attribute__((address_space(1))) v4i_vs* gas_v4i;
typedef __attribute__((address_space(3))) v4i_vs* lds_v4i;

__device__ __forceinline__ lds_v4i to_lds_v4i(void* p) {
    // generic LDS pointer: low 32 bits are the LDS byte offset
    return (lds_v4i)(unsigned int)(unsigned long long)p;
}
__device__ __forceinline__ gas_v4i to_gas_v4i(const void* p) {
    return (gas_v4i)(unsigned long long)p;
}

// copy 16B global -> LDS (async DMA if available, else load+store)
__device__ __forceinline__ void cp16(void* ldst, const void* gsrc) {
#if defined(USE_ASYNC_LDS)
    __builtin_amdgcn_global_load_async_to_lds_b128(to_gas_v4i(gsrc), to_lds_v4i(ldst), 0, 0);
#else
    *(float4*)ldst = *(const float4*)gsrc;
#endif
}
// wait until <= 8 asyncs in flight (drains previous slab, keeps next in flight)
__device__ __forceinline__ void cp_wait8() {
#if defined(USE_ASYNC_LDS)
    __builtin_amdgcn_s_wait_asynccnt(8);
#endif
}
__device__ __forceinline__ void cp_wait0() {
#if defined(USE_ASYNC_LDS)
    __builtin_amdgcn_s_wait_asynccnt(0);
#endif
}

// ======================================================================
// GroupNorm: x (B, C_, L) -> out, 32 groups.  Optional fused combine:
// out = (addIn + norm)*0.5 (used for lce).  In-place safe (stats first).
// ======================================================================
__global__ __launch_bounds__(256)
void gn_kernel(const float* __restrict__ x,
               const float* __restrict__ gamma,
               const float* __restrict__ beta,
               const float* __restrict__ addIn,
               float* __restrict__ out,
               int C_, int L)
{
    const int blk = blockIdx.x;
    const int b   = blk / GROUPS;
    const int g   = blk % GROUPS;
    const int cg  = C_ / GROUPS;
    const int n   = cg * L;
    const size_t base = ((size_t)b * C_ + (size_t)g * cg) * L;
    const float* xp = x + base;

    float s = 0.f, s2 = 0.f;
    for (int i = threadIdx.x; i < n; i += blockDim.x) {
        float v = xp[i];
        s += v; s2 += v * v;
    }
    __shared__ float r1[256], r2[256];
    r1[threadIdx.x] = s; r2[threadIdx.x] = s2;
    __syncthreads();
    for (int off = 128; off > 0; off >>= 1) {
        if ((int)threadIdx.x < off) {
            r1[threadIdx.x] += r1[threadIdx.x + off];
            r2[threadIdx.x] += r2[threadIdx.x + off];
        }
        __syncthreads();
    }
    const float mean = r1[0] / (float)n;
    const float var  = r2[0] / (float)n - mean * mean;
    const float inv  = rsqrtf(var + EPSV);

    const float* ap = addIn ? (addIn + base) : nullptr;
    float* op = out + base;
    for (int i = threadIdx.x; i < n; i += blockDim.x) {
        const int ch = g * cg + i / L;
        float v = (xp[i] - mean) * inv * gamma[ch] + beta[ch];
        if (ap) v = (ap[i] + v) * 0.5f;
        op[i] = v;
    }
}

// ======================================================================
// Batched GEMM via fp32 WMMA, double-buffered async-LDS pipeline:
//   OUT[b] = W (MxK) * X[b] (KxN) + bias  (+ optional residual)
// Block = 128 threads (4 waves) computes a 64x64 tile.  K-slabs of W/X
// ping-pong between two LDS buffers: slab k+1 is DMA'd (async-to-LDS)
// while slab k feeds the WMMAs; s_wait_asynccnt 8 drains only the
// previous slab (async completion is in-order).  grid = (N/64, M/64, B).
// ======================================================================
__global__ __launch_bounds__(128)
void gemm_wmma(const float* __restrict__ W,
               const float* __restrict__ X,
               const float* __restrict__ bias,
               const float* __restrict__ residual,
               float* __restrict__ OUT,
               int M, int K, int N)
{
    const int tid  = threadIdx.x;
    const int w    = tid >> 5;
    const int lane = tid & 31;
    const int l16  = lane & 15;
    const int half = lane >> 4;
    const int n0 = blockIdx.x * 64;
    const int m0 = blockIdx.y * 64;
    const int b  = blockIdx.z;

    // +4 pad: rows stay 16B-aligned for b128 copies, banks spread for reads
    __shared__ float W_s[2][64][KC + 4];
    __shared__ float X_s[2][KC][64 + 4];

    const float* Xb = X + (size_t)b * K * N;

    // stage one K-slab into buffer `buf` (8 async b128 per wave: 4 W + 4 X)
    auto stage = [&](int buf, int kb) {
        for (int i = tid; i < 64 * (KC / 4); i += 128) {
            const int r  = i >> 3;
            const int kq = (i & 7) * 4;
            cp16(&W_s[buf][r][kq], W + (size_t)(m0 + r) * K + kb + kq);
        }
        for (int i = tid; i < KC * 16; i += 128) {
            const int r  = i >> 4;
            const int nq = (i & 15) * 4;
            cp16(&X_s[buf][r][nq], Xb + (size_t)(kb + r) * N + n0 + nq);
        }
    };

    v8f acc[4];
#pragma unroll
    for (int nt = 0; nt < 4; ++nt)
#pragma unroll
        for (int i = 0; i < 8; ++i) acc[nt][i] = 0.f;

    stage(0, 0);                       // prologue: slab 0 in flight
    int cur = 0;
    for (int kb = 0; kb < K; kb += KC) {
        __syncthreads();               // WAR: all waves done reading buf cur^1
        if (kb + KC < K) {
            stage(cur ^ 1, kb + KC);   // overlap next slab's DMA with compute
            cp_wait8();                // drain slab `cur` only (in-order)
        } else {
            cp_wait0();
        }
        __syncthreads();               // slab `cur` visible to all waves

#pragma unroll
        for (int k0 = 0; k0 < KC; k0 += 4) {
            const int ka = k0 + 2 * half;   // A: V0 -> K=k0/k0+2, V1 -> k0+1/k0+3
            v2f a;
            a[0] = W_s[cur][w * 16 + l16][ka];
            a[1] = W_s[cur][w * 16 + l16][ka + 1];
#pragma unroll
            for (int nt = 0; nt < 4; ++nt) {
                v2f bb;
                bb[0] = X_s[cur][ka    ][nt * 16 + l16];
                bb[1] = X_s[cur][ka + 1][nt * 16 + l16];
                acc[nt] = __builtin_amdgcn_wmma_f32_16x16x4_f32(
                              false, a, false, bb, (short)0, acc[nt], false, false);
            }
        }
        cur ^= 1;
    }

    float* Ob = OUT + (size_t)b * M * N;
#pragma unroll
    for (int nt = 0; nt < 4; ++nt)
#pragma unroll
        for (int i = 0; i < 8; ++i) {       // D: VGPR i -> row +i (half0) / +8+i
            const int m = m0 + w * 16 + i + 8 * half;
            const int n = n0 + nt * 16 + l16;
            float v = acc[nt][i] + bias[m];
            if (residual) v += residual[(size_t)b * M * N + (size_t)m * N + n];
            Ob[(size_t)m * N + n] = v;
        }
}

// ======================================================================
// Fused flash attention (online softmax).  Block = 64 threads = 2 waves;
// each wave owns 16 query rows, block covers 32 queries of one (b, head).
// k_mix/v_mix gathered on the fly from qkv / ipp / klvl / lpe.
// ======================================================================
__global__ __launch_bounds__(64)
void attn_kernel(const float* __restrict__ qkv,
                 const float* __restrict__ ipp,
                 const float* __restrict__ klvl,
                 const float* __restrict__ lpe,
                 const unsigned char* __restrict__ kpm,
                 float* __restrict__ attout)
{
    const int b   = blockIdx.z;
    const int h   = blockIdx.y;
    const int tq0 = blockIdx.x * 32;
    const int w    = threadIdx.x >> 5;
    const int lane = threadIdx.x & 31;
    const int l16  = lane & 15;
    const int half = lane >> 4;
    const float sc = 0.08838834764831845f; // 1/sqrt(128) = scale*scale

    __shared__ float q_s[128][33];
    __shared__ float k_s[128][17];
    __shared__ float v_s[64][17];
    __shared__ float p_s[2][16][17];

    const float* qb  = qkv  + ((size_t)b * 3 * CC +            h * DHEAD) * L1C;
    const float* kb  = qkv  + ((size_t)b * 3 * CC + CC   +     h * DHEAD) * L1C;
    const float* vb  = qkv  + ((size_t)b * 3 * CC + 2*CC +     h * DHEAD) * L1C;
    const float* ipb = ipp  + ((size_t)b * CC  + h * DHEAD) * L1C;
    const float* klb = klvl + ((size_t)b * 2*CC +        h * DHEAD) * L2C;
    const float* vlb = klvl + ((size_t)b * 2*CC + CC +   h * DHEAD) * L2C;
    const float* lpb = lpe  + ((size_t)b * CC  + h * DHEAD) * L2C;

    for (int i = threadIdx.x; i < 128 * 32; i += 64) {
        const int c = i >> 5, t = i & 31;
        q_s[c][t] = (c < DHEAD)
                  ? qb [(size_t)c          * L1C + tq0 + t]
                  : ipb[(size_t)(c - DHEAD)* L1C + tq0 + t];
    }

    v8f accO[4];
#pragma unroll
    for (int nt = 0; nt < 4; ++nt)
#pragma unroll
        for (int i = 0; i < 8; ++i) accO[nt][i] = 0.f;
    float m8[8], l8[8];
#pragma unroll
    for (int i = 0; i < 8; ++i) { m8[i] = -1e30f; l8[i] = 0.f; }

    for (int s0 = 0; s0 < L1C + L2C; s0 += 16) {
        // prefetch next key/value tile rows (global_prefetch_b8)
        if (s0 + 16 < L1C) {
            const int c = threadIdx.x;               // 0..63
            __builtin_prefetch(kb  + (size_t)c * L1C + s0 + 16, 0, 1);
            __builtin_prefetch(ipb + (size_t)c * L1C + s0 + 16, 0, 1);
            __builtin_prefetch(vb  + (size_t)c * L1C + s0 + 16, 0, 1);
        }
        __syncthreads();
        for (int i = threadIdx.x; i < 128 * 16; i += 64) {
            const int c = i >> 4, sl = i & 15, s = s0 + sl;
            float v;
            if (c < DHEAD)
                v = (s < L1C) ? kb [(size_t)c * L1C + s]
                              : klb[(size_t)c * L2C + (s - L1C)];
            else
                v = (s < L1C) ? ipb[(size_t)(c - DHEAD) * L1C + s]
                              : lpb[(size_t)(c - DHEAD) * L2C + (s - L1C)];
            k_s[c][sl] = v;
        }
        for (int i = threadIdx.x; i < 64 * 16; i += 64) {
            const int c = i >> 4, sl = i & 15, s = s0 + sl;
            v_s[c][sl] = (s < L1C) ? vb [(size_t)c * L1C + s]
                                   : vlb[(size_t)c * L2C + (s - L1C)];
        }
        __syncthreads();

        // S = (Q tile)^T (K tile) : contract over 128 channels
        v8f S = {};
        for (int k0 = 0; k0 < 128; k0 += 4) {
            const int ka = k0 + 2 * half;
            v2f a, bb;
            a[0]  = q_s[ka    ][w * 16 + l16];
            a[1]  = q_s[ka + 1][w * 16 + l16];
            bb[0] = k_s[ka    ][l16];
            bb[1] = k_s[ka + 1][l16];
            S = __builtin_amdgcn_wmma_f32_16x16x4_f32(
                    false, a, false, bb, (short)0, S, false, false);
        }

        const int sg = s0 + l16;
        const bool masked = (sg >= L1C) && (kpm[b * L2C + (sg - L1C)] != 0);
        float alpha8[8];
#pragma unroll
        for (int i = 0; i < 8; ++i) {
            float logit = masked ? -1e30f : S[i] * sc;
            float tm = logit;                     // row max over 16 lanes of half
            tm = fmaxf(tm, __shfl_xor(tm, 1, 32));
            tm = fmaxf(tm, __shfl_xor(tm, 2, 32));
            tm = fmaxf(tm, __shfl_xor(tm, 4, 32));
            tm = fmaxf(tm, __shfl_xor(tm, 8, 32));
            const float mnew  = fmaxf(m8[i], tm);
            const float alpha = __expf(m8[i] - mnew);
            const float p     = __expf(logit - mnew);
            float ts = p;
            ts += __shfl_xor(ts, 1, 32);
            ts += __shfl_xor(ts, 2, 32);
            ts += __shfl_xor(ts, 4, 32);
            ts += __shfl_xor(ts, 8, 32);
            l8[i] = l8[i] * alpha + ts;
            m8[i] = mnew;
            alpha8[i] = alpha;
            p_s[w][i + 8 * half][l16] = p;
        }
#pragma unroll
        for (int nt = 0; nt < 4; ++nt)
#pragma unroll
            for (int i = 0; i < 8; ++i) accO[nt][i] *= alpha8[i];

        __syncthreads();
        // O += P (16x16) * V^T (16 x 64)
#pragma unroll
        for (int nt = 0; nt < 4; ++nt) {
            v8f acc = accO[nt];
            for (int k0 = 0; k0 < 16; k0 += 4) {
                const int ka = k0 + 2 * half;
                v2f a, bb;
                a[0]  = p_s[w][l16][ka];
                a[1]  = p_s[w][l16][ka + 1];
                bb[0] = v_s[nt * 16 + l16][ka];
                bb[1] = v_s[nt * 16 + l16][ka + 1];
                acc = __builtin_amdgcn_wmma_f32_16x16x4_f32(
                          false, a, false, bb, (short)0, acc, false, false);
            }
            accO[nt] = acc;
        }
    }

#pragma unroll
    for (int nt = 0; nt < 4; ++nt)
#pragma unroll
        for (int i = 0; i < 8; ++i) {
            const int t = tq0 + w * 16 + i + 8 * half;
            const int c = nt * 16 + l16;
            attout[((size_t)b * CC + h * DHEAD + c) * L1C + t] = accO[nt][i] / l8[i];
        }
}

// ======================================================================
// host-side launcher
// ======================================================================
extern "C" void kernel_launch(void* const* d_in, const int* in_sizes, int n_in,
                              void* d_out, int out_size, void* d_ws, size_t ws_size,
                              hipStream_t stream) {
    (void)in_sizes; (void)n_in; (void)out_size; (void)ws_size;

    const float* x        = (const float*)d_in[0];
    const float* xf_out   = (const float*)d_in[1];
    const float* obj_cls  = (const float*)d_in[2];
    const float* obj_bbox = (const float*)d_in[3];
    const float* img_bbox = (const float*)d_in[4];
    const unsigned char* kpm = (const unsigned char*)d_in[5];
    const float* qkv_w  = (const float*)d_in[6];
    const float* qkv_b  = (const float*)d_in[7];
    const float* g_qkv  = (const float*)d_in[8];
    const float* b_qkv  = (const float*)d_in[9];
    const float* lc_w   = (const float*)d_in[10];
    const float* lc_b   = (const float*)d_in[11];
    const float* lp_w   = (const float*)d_in[12];
    const float* lp_b   = (const float*)d_in[13];
    const float* g_obj  = (const float*)d_in[14];
    const float* b_obj  = (const float*)d_in[15];
    const float* g_lpos = (const float*)d_in[16];
    const float* b_lpos = (const float*)d_in[17];
    const float* g_ipos = (const float*)d_in[18];
    const float* b_ipos = (const float*)d_in[19];
    const float* proj_w = (const float*)d_in[20];
    const float* proj_b = (const float*)d_in[21];
    float* out = (float*)d_out;

    // workspace layout (floats)
    float* ws   = (float*)d_ws;
    float* xn   = ws;                         // B*C*L1      = 4,194,304
    float* qkv  = xn   + (size_t)BB*CC*L1C;   // B*3C*L1     = 12,582,912
    float* ipp  = qkv  + (size_t)BB*3*CC*L1C; // B*C*L1      = 4,194,304
    float* lpe  = ipp  + (size_t)BB*CC*L1C;   // B*C*L2      = 262,144
    float* lce  = lpe  + (size_t)BB*CC*L2C;   // B*ENC*L2    = 524,288
    float* klvl = lce  + (size_t)BB*ENC*L2C;  // B*2C*L2     = 524,288
    float* att  = klvl + (size_t)BB*2*CC*L2C; // B*C*L1      = 4,194,304

    // 1) xn = GroupNorm(x)
    gn_kernel<<<BB * GROUPS, 256, 0, stream>>>(x, g_qkv, b_qkv, nullptr, xn, CC, L1C);

    // 2) qkv = qkv_w (1536x512) * xn (512x1024) + qkv_b
    gemm_wmma<<<dim3(L1C/64, 3*CC/64, BB), 128, 0, stream>>>(
        qkv_w, xn, qkv_b, nullptr, qkv, 3*CC, CC, L1C);

    // 3) ipp = GN(lp_w (512x1024) * img_bbox (1024x1024) + lp_b)
    gemm_wmma<<<dim3(L1C/64, CC/64, BB), 128, 0, stream>>>(
        lp_w, img_bbox, lp_b, nullptr, ipp, CC, ENC, L1C);
    gn_kernel<<<BB * GROUPS, 256, 0, stream>>>(ipp, g_ipos, b_ipos, nullptr, ipp, CC, L1C);

    // 4) lpe = GN(lp_w * obj_bbox (1024x64) + lp_b)
    gemm_wmma<<<dim3(L2C/64, CC/64, BB), 128, 0, stream>>>(
        lp_w, obj_bbox, lp_b, nullptr, lpe, CC, ENC, L2C);
    gn_kernel<<<BB * GROUPS, 256, 0, stream>>>(lpe, g_lpos, b_lpos, nullptr, lpe, CC, L2C);

    // 5) lce = (xf_out + GN(obj_class)) / 2
    gn_kernel<<<BB * GROUPS, 256, 0, stream>>>(obj_cls, g_obj, b_obj, xf_out, lce, ENC, L2C);

    // 6) klvl = lc_w (1024x1024) * lce (1024x64) + lc_b
    gemm_wmma<<<dim3(L2C/64, 2*CC/64, BB), 128, 0, stream>>>(
        lc_w, lce, lc_b, nullptr, klvl, 2*CC, ENC, L2C);

    // 7) fused flash attention -> att (B, C, L1)
    attn_kernel<<<dim3(L1C/32, NHEAD, BB), 64, 0, stream>>>(
        qkv, ipp, klvl, lpe, kpm, att);

    // 8) out = x + proj_w (512x512) * att + proj_b
    gemm_wmma<<<dim3(L1C/64, CC/64, BB), 128, 0, stream>>>(
        proj_w, att, proj_b, x, out, CC, CC, L1C);
}